// NBST_79766132621711
// MI455X (gfx1250) — compile-verified
//
#include <hip/hip_runtime.h>

// ---------------------------------------------------------------------------
// NBST forward for MI455X (gfx1250, wave32).
// f32 WMMA (v_wmma_f32_16x16x4_f32) for all GEMMs; two-layer MLP + LayerNorm
// fused per 16-row tile with hidden activations resident in LDS, and the
// input gather/concat fused into the A-tile staging (no X intermediates in
// HBM -> saves ~570 MB of round-trip traffic on a bandwidth-bound problem).
// ---------------------------------------------------------------------------

typedef __attribute__((ext_vector_type(2))) float v2f;
typedef __attribute__((ext_vector_type(8))) float v8f;

#define BB 32
#define TT 168
#define NN 50
#define HH 128
#define EE 8
#define EPSF 1e-5f

// ---------------- activations ----------------
template <int ACT>
__device__ __forceinline__ float actf(float x) {
  if (ACT == 0) return 1.f / (1.f + expf(-x));   // sigmoid
  if (ACT == 1) return tanhf(x);                 // tanh
  return fmaxf(x, 0.f);                          // relu
}

// ---------------- fused input gather ----------------
struct GP {
  const float* f0;
  const float* f1;
  const float* f2;
  const float* f3;
  const int*   e0;
  const int*   e1;
};

// MODE 0: static  -> [station_nodes | local_node], K=16
// MODE 1: dynamic -> [features(11) | weather_emb(8) | wind_emb(8) | pad], K=28(27)
// MODE 2: fusion  -> [st_emb(128) | sf_emb(128) | pm25_emb(8)], K=264
template <int MODE>
__device__ __forceinline__ float gatherA(const GP& g, int row, int c) {
  if (MODE == 0) {
    return (row < BB * NN) ? g.f0[row * 16 + c]
                           : g.f1[(row - BB * NN) * 16 + c];
  } else if (MODE == 1) {
    const int M1 = BB * TT * NN;
    if (row < M1) {
      if (c < 11)  return g.f0[row * 11 + c];
      if (c < 19)  return g.f2[g.e0[row * 6 + 1] * EE + (c - 11)];
      if (c < 27)  return g.f3[g.e0[row * 6 + 2] * EE + (c - 19)];
      return 0.f;
    } else {
      int lr = row - M1;
      if (c < 11)  return g.f1[lr * 11 + c];
      if (c < 19)  return g.f2[g.e1[lr * 2 + 0] * EE + (c - 11)];
      if (c < 27)  return g.f3[g.e1[lr * 2 + 1] * EE + (c - 19)];
      return 0.f;
    }
  } else {
    if (c < 128) {
      int n = row % NN;
      int b = row / (TT * NN);
      return g.f0[(b * NN + n) * 128 + c];
    } else if (c < 256) {
      return g.f1[(size_t)row * 128 + (c - 128)];
    } else {
      return g.f2[g.e0[row * 6 + 0] * EE + (c - 256)];
    }
  }
}

// ---------------------------------------------------------------------------
// Fused gather + 2-layer MLP + LayerNorm.
// out[M,128] = LN(act(act(A@W1+b1)@W2+b2)),  A gathered on the fly.
// One block = 16 rows; 128 threads = 4 waves; 16x16 f32 WMMA tiles.
// ---------------------------------------------------------------------------
template <int ACT, int MODE, int KPAD, int KACT>
__global__ void fused_mlp_kernel(GP g, int M,
                                 const float* __restrict__ W1, const float* __restrict__ b1,
                                 const float* __restrict__ W2, const float* __restrict__ b2,
                                 const float* __restrict__ lnw, const float* __restrict__ lnb,
                                 float* __restrict__ out) {
  __shared__ float As[16 * KPAD];
  __shared__ float H1s[16 * 256];
  __shared__ float H2s[16 * 128];
  __shared__ float rmean[16], rinv[16];

  const int tid  = threadIdx.x;
  const int wave = tid >> 5;
  const int lane = tid & 31;
  const int mn   = lane & 15;     // A row / B col / D col for this lane
  const int half = lane >> 4;
  const int row0 = blockIdx.x * 16;

  // gather A tile straight into LDS (no HBM intermediate)
  for (int i = tid; i < 16 * KPAD; i += 128) {
    int r = i / KPAD, c = i - r * KPAD;
    int row = row0 + r;
    As[i] = (row < M) ? gatherA<MODE>(g, row, c) : 0.f;
  }
  __syncthreads();

  // GEMM1: [16,KPAD] x [KACT,256] -> H1s, activation fused
  for (int ct = wave; ct < 16; ct += 4) {
    int col = ct * 16 + mn;
    v8f acc = {};
#pragma unroll
    for (int k0 = 0; k0 < KPAD; k0 += 4) {
      int ka = k0 + half * 2;
      v2f a, bv;
      a.x  = As[mn * KPAD + ka];
      a.y  = As[mn * KPAD + ka + 1];
      bv.x = (ka < KACT)     ? W1[(size_t)ka * 256 + col]       : 0.f;
      bv.y = (ka + 1 < KACT) ? W1[(size_t)(ka + 1) * 256 + col] : 0.f;
      acc = __builtin_amdgcn_wmma_f32_16x16x4_f32(false, a, false, bv, (short)0, acc,
                                                  false, false);
    }
    float bb = b1[col];
#pragma unroll
    for (int v = 0; v < 8; ++v) {
      int r = v + 8 * half;                  // D row per ISA C/D layout
      H1s[r * 256 + col] = actf<ACT>(acc[v] + bb);
    }
  }
  __syncthreads();

  // GEMM2: [16,256] x [256,128] -> H2s, activation fused
  for (int ct = wave; ct < 8; ct += 4) {
    int col = ct * 16 + mn;
    v8f acc = {};
    for (int k0 = 0; k0 < 256; k0 += 4) {
      int ka = k0 + half * 2;
      v2f a, bv;
      a.x  = H1s[mn * 256 + ka];
      a.y  = H1s[mn * 256 + ka + 1];
      bv.x = W2[(size_t)ka * 128 + col];
      bv.y = W2[(size_t)(ka + 1) * 128 + col];
      acc = __builtin_amdgcn_wmma_f32_16x16x4_f32(false, a, false, bv, (short)0, acc,
                                                  false, false);
    }
    float bb = b2[col];
#pragma unroll
    for (int v = 0; v < 8; ++v) {
      int r = v + 8 * half;
      H2s[r * 128 + col] = actf<ACT>(acc[v] + bb);
    }
  }
  __syncthreads();

  // LayerNorm over last dim (128)
  if (tid < 16) {
    float s = 0.f, sq = 0.f;
    for (int c = 0; c < 128; ++c) {
      float x = H2s[tid * 128 + c];
      s += x; sq += x * x;
    }
    float m = s * (1.f / 128.f);
    float v = sq * (1.f / 128.f) - m * m;
    rmean[tid] = m;
    rinv[tid]  = rsqrtf(v + EPSF);
  }
  __syncthreads();

  for (int i = tid; i < 16 * 128; i += 128) {
    int r = i >> 7, c = i & 127;
    int row = row0 + r;
    if (row < M)
      out[(size_t)row * 128 + c] = (H2s[i] - rmean[r]) * rinv[r] * lnw[c] + lnb[c];
  }
}

// ---------------------------------------------------------------------------
// Plain GEMM + bias (GRU input projection): out[M,Nc] = X@W + bias.
// K multiple of 4, Nc multiple of 16.
// ---------------------------------------------------------------------------
__global__ void gemm_bias_kernel(const float* __restrict__ X, int M, int K,
                                 const float* __restrict__ W, const float* __restrict__ bias,
                                 int Nc, float* __restrict__ out) {
  __shared__ float As[16 * 536];   // K = 536 here
  const int tid  = threadIdx.x;
  const int wave = tid >> 5;
  const int lane = tid & 31;
  const int mn   = lane & 15;
  const int half = lane >> 4;
  const int row0 = blockIdx.x * 16;

  for (int i = tid; i < 16 * K; i += 128) {
    int r = i / K, c = i - r * K;
    int row = row0 + r;
    As[i] = (row < M) ? X[(size_t)row * K + c] : 0.f;
  }
  __syncthreads();

  const int ntiles = Nc / 16;
  for (int ct = wave; ct < ntiles; ct += 4) {
    int col = ct * 16 + mn;
    v8f acc = {};
    for (int k0 = 0; k0 < K; k0 += 4) {
      int ka = k0 + half * 2;
      v2f a, bv;
      a.x  = As[mn * K + ka];
      a.y  = As[mn * K + ka + 1];
      bv.x = W[(size_t)ka * Nc + col];
      bv.y = W[(size_t)(ka + 1) * Nc + col];
      acc = __builtin_amdgcn_wmma_f32_16x16x4_f32(false, a, false, bv, (short)0, acc,
                                                  false, false);
    }
    float bb = bias[col];
#pragma unroll
    for (int v = 0; v < 8; ++v) {
      int r = v + 8 * half;
      int row = row0 + r;
      if (row < M) out[(size_t)row * Nc + col] = acc[v] + bb;
    }
  }
}

// ---------------------------------------------------------------------------
// Attention (dot-product over 50 stations + softmax + weighted sum).
// One block per query row g.  Used for both static (g=b) and dynamic (g=b*T+t).
// ---------------------------------------------------------------------------
__global__ void attn_kernel(const float* __restrict__ semb,   // [G*50, 128]
                            const float* __restrict__ lemb,   // [G, 128]
                            float* __restrict__ attn,         // [G, 50]
                            float* __restrict__ aout) {       // [G, 128]
  __shared__ float sc[64];
  __shared__ float sinv;
  int g = blockIdx.x, tid = threadIdx.x;
  const float scale = rsqrtf((float)NN);
  if (tid < NN) {
    float d = 0.f;
    for (int h = 0; h < 128; ++h)
      d += lemb[g * 128 + h] * semb[(size_t)(g * NN + tid) * 128 + h];
    sc[tid] = d * scale;
  }
  __syncthreads();
  if (tid == 0) {
    float mx = sc[0];
    for (int i = 1; i < NN; ++i) mx = fmaxf(mx, sc[i]);
    float s = 0.f;
    for (int i = 0; i < NN; ++i) { sc[i] = expf(sc[i] - mx); s += sc[i]; }
    sinv = 1.f / s;
  }
  __syncthreads();
  if (tid < NN) { sc[tid] *= sinv; attn[g * NN + tid] = sc[tid]; }
  __syncthreads();
  float acc = 0.f;
  for (int n = 0; n < NN; ++n)
    acc += sc[n] * semb[(size_t)(g * NN + n) * 128 + tid];
  aout[g * 128 + tid] = acc;
}

// ---------------- batch-norm (training-mode) over rows, channels last ------
__global__ void bn_stats_kernel(const float* __restrict__ x, int rows,
                                float* __restrict__ stats /* [256]: mean, inv */) {
  int h = threadIdx.x;
  float s = 0.f, sq = 0.f;
  for (int r = 0; r < rows; ++r) {
    float v = x[(size_t)r * 128 + h];
    s += v; sq += v * v;
  }
  float m = s / (float)rows;
  float var = sq / (float)rows - m * m;
  stats[h]       = m;
  stats[128 + h] = rsqrtf(var + EPSF);
}

__global__ void bn_apply_kernel(const float* __restrict__ x,
                                const float* __restrict__ stats,
                                const float* __restrict__ w, const float* __restrict__ b,
                                float* __restrict__ out) {
  int r = blockIdx.x, h = threadIdx.x;
  out[(size_t)r * 128 + h] =
      (x[(size_t)r * 128 + h] - stats[h]) * stats[128 + h] * w[h] + b[h];
}

// --------- stat_att / dyn_att: weighted sums of stations_feat over N -------
__global__ void att_combine_kernel(const float* __restrict__ sfeat,  // [BT*50,128]
                                   const float* __restrict__ attn_s, // [B,50]
                                   const float* __restrict__ attn_d, // [BT,50]
                                   float* __restrict__ stat_att,     // [BT,128]
                                   float* __restrict__ dyn_att) {    // [BT,128]
  int bt = blockIdx.x, h = threadIdx.x;
  int b = bt / TT;
  float sa = 0.f, da = 0.f;
  for (int n = 0; n < NN; ++n) {
    float f = sfeat[(size_t)(bt * NN + n) * 128 + h];
    sa += attn_s[b * NN + n] * f;
    da += attn_d[bt * NN + n] * f;
  }
  stat_att[(size_t)bt * 128 + h] = sa;
  dyn_att[(size_t)bt * 128 + h]  = da;
}

// ---------------- tf concat: [B*T, 536] ------------------------------------
__global__ void build_tf_kernel(const float* __restrict__ static_norm, // [B,128]
                                const float* __restrict__ dyn_norm,    // [BT,128]
                                const float* __restrict__ stat_att,    // [BT,128]
                                const float* __restrict__ dyn_att,     // [BT,128]
                                const int*   __restrict__ station_emb,
                                const float* __restrict__ month_emb,
                                const float* __restrict__ day_emb,
                                const float* __restrict__ hour_emb,
                                float* __restrict__ tf) {
  int idx = blockIdx.x * 128 + threadIdx.x;
  const int total = BB * TT * 536;
  if (idx >= total) return;
  int row = idx / 536, c = idx - row * 536;
  int b = row / TT;
  int e0 = row * NN * 6;   // station (b,t,n=0)
  float v;
  if (c < 128)       v = static_norm[b * 128 + c];
  else if (c < 256)  v = dyn_norm[(size_t)row * 128 + (c - 128)];
  else if (c < 384)  v = stat_att[(size_t)row * 128 + (c - 256)];
  else if (c < 512)  v = dyn_att[(size_t)row * 128 + (c - 384)];
  else if (c < 520)  v = month_emb[(station_emb[e0 + 3] - 1) * EE + (c - 512)];
  else if (c < 528)  v = day_emb[station_emb[e0 + 4] * EE + (c - 520)];
  else               v = hour_emb[station_emb[e0 + 5] * EE + (c - 528)];
  tf[idx] = v;
}

__global__ void transpose_kernel(const float* __restrict__ W /*384x536*/,
                                 float* __restrict__ WT /*536x384*/) {
  int idx = blockIdx.x * 128 + threadIdx.x;
  if (idx >= 384 * 536) return;
  int j = idx / 536, k = idx - j * 536;
  WT[k * 384 + j] = W[idx];
}

// ---------------------------------------------------------------------------
// 2-layer GRU over T, one block per batch element, one hidden unit per thread.
// gi0 (= tf @ Wih0^T + bih0) is precomputed.  Prediction head fused.
// ---------------------------------------------------------------------------
__global__ void gru_kernel(const float* __restrict__ gi0,   // [BT, 384]
                           const float* __restrict__ Whh0, const float* __restrict__ bhh0,
                           const float* __restrict__ Wih1, const float* __restrict__ bih1,
                           const float* __restrict__ Whh1, const float* __restrict__ bhh1,
                           const float* __restrict__ predW, const float* __restrict__ predb,
                           float* __restrict__ out) {       // [BT]
  __shared__ float h1[128], h2[128], red[128];
  int b = blockIdx.x, tid = threadIdx.x;
  h1[tid] = 0.f; h2[tid] = 0.f;
  __syncthreads();

  for (int t = 0; t < TT; ++t) {
    int row = b * TT + t;
    // layer 1
    float gir = gi0[(size_t)row * 384 + tid];
    float giz = gi0[(size_t)row * 384 + 128 + tid];
    float gic = gi0[(size_t)row * 384 + 256 + tid];
    float ghr = bhh0[tid], ghz = bhh0[128 + tid], ghc = bhh0[256 + tid];
    for (int k = 0; k < 128; ++k) {
      float hv = h1[k];
      ghr += Whh0[tid * 128 + k] * hv;
      ghz += Whh0[(128 + tid) * 128 + k] * hv;
      ghc += Whh0[(256 + tid) * 128 + k] * hv;
    }
    float r = 1.f / (1.f + expf(-(gir + ghr)));
    float z = 1.f / (1.f + expf(-(giz + ghz)));
    float n = tanhf(gic + r * ghc);
    float h1new = (1.f - z) * n + z * h1[tid];
    __syncthreads(); h1[tid] = h1new; __syncthreads();

    // layer 2
    float g2r = bih1[tid], g2z = bih1[128 + tid], g2c = bih1[256 + tid];
    float q2r = bhh1[tid], q2z = bhh1[128 + tid], q2c = bhh1[256 + tid];
    for (int k = 0; k < 128; ++k) {
      float hv = h1[k], h2v = h2[k];
      g2r += Wih1[tid * 128 + k] * hv;
      g2z += Wih1[(128 + tid) * 128 + k] * hv;
      g2c += Wih1[(256 + tid) * 128 + k] * hv;
      q2r += Whh1[tid * 128 + k] * h2v;
      q2z += Whh1[(128 + tid) * 128 + k] * h2v;
      q2c += Whh1[(256 + tid) * 128 + k] * h2v;
    }
    float r2 = 1.f / (1.f + expf(-(g2r + q2r)));
    float z2 = 1.f / (1.f + expf(-(g2z + q2z)));
    float n2 = tanhf(g2c + r2 * q2c);
    float h2new = (1.f - z2) * n2 + z2 * h2[tid];
    __syncthreads(); h2[tid] = h2new; __syncthreads();

    // prediction head
    red[tid] = h2new * predW[tid];
    __syncthreads();
    if (tid == 0) {
      float s = 0.f;
      for (int k = 0; k < 128; ++k) s += red[k];
      out[row] = tanhf(s + predb[0]);
    }
    __syncthreads();
  }
}

// ---------------------------------------------------------------------------
extern "C" void kernel_launch(void* const* d_in, const int* in_sizes, int n_in,
                              void* d_out, int out_size, void* d_ws, size_t ws_size,
                              hipStream_t stream) {
  (void)in_sizes; (void)n_in; (void)out_size; (void)ws_size;

  // ---- inputs (setup_inputs dict order, params in insertion order) ----
  const float* local_node       = (const float*)d_in[0];
  const float* local_features   = (const float*)d_in[1];
  const int*   local_emb        = (const int*)  d_in[2];
  const float* station_nodes    = (const float*)d_in[3];
  const float* station_features = (const float*)d_in[4];
  const int*   station_emb      = (const int*)  d_in[5];
  const float* pm25_emb    = (const float*)d_in[6];
  const float* weather_emb = (const float*)d_in[7];
  const float* wind_emb    = (const float*)d_in[8];
  const float* month_emb   = (const float*)d_in[9];
  const float* day_emb     = (const float*)d_in[10];
  const float* hour_emb    = (const float*)d_in[11];
  const float* st_W1 = (const float*)d_in[12]; const float* st_b1 = (const float*)d_in[13];
  const float* st_W2 = (const float*)d_in[14]; const float* st_b2 = (const float*)d_in[15];
  const float* st_lnw = (const float*)d_in[16]; const float* st_lnb = (const float*)d_in[17];
  const float* bn_s_w = (const float*)d_in[18]; const float* bn_s_b = (const float*)d_in[19];
  const float* dy_W1 = (const float*)d_in[20]; const float* dy_b1 = (const float*)d_in[21];
  const float* dy_W2 = (const float*)d_in[22]; const float* dy_b2 = (const float*)d_in[23];
  const float* dy_lnw = (const float*)d_in[24]; const float* dy_lnb = (const float*)d_in[25];
  const float* bn_d_w = (const float*)d_in[26]; const float* bn_d_b = (const float*)d_in[27];
  const float* sf_W1 = (const float*)d_in[28]; const float* sf_b1 = (const float*)d_in[29];
  const float* sf_W2 = (const float*)d_in[30]; const float* sf_b2 = (const float*)d_in[31];
  const float* sf_lnw = (const float*)d_in[32]; const float* sf_lnb = (const float*)d_in[33];
  const float* gru_Wih0 = (const float*)d_in[34]; const float* gru_Whh0 = (const float*)d_in[35];
  const float* gru_bih0 = (const float*)d_in[36]; const float* gru_bhh0 = (const float*)d_in[37];
  const float* gru_Wih1 = (const float*)d_in[38]; const float* gru_Whh1 = (const float*)d_in[39];
  const float* gru_bih1 = (const float*)d_in[40]; const float* gru_bhh1 = (const float*)d_in[41];
  const float* pred_W = (const float*)d_in[42]; const float* pred_b = (const float*)d_in[43];

  float* out = (float*)d_out;

  // ---- workspace layout (floats) ----
  const int M1 = BB * TT * NN;       // 268800 station rows
  const int BT = BB * TT;            // 5376
  const int M_st = BB * NN + BB;     // 1632
  const int M_dy = M1 + BT;          // 274176

  float* p = (float*)d_ws;
  auto take = [&](size_t n) { float* r = p; p += n; return r; };
  float* st_out      = take((size_t)M_st * 128);    // [0,1600): station emb, [1600,1632): local
  float* attn_s      = take((size_t)BB * NN);
  float* static_out  = take((size_t)BB * 128);
  float* static_norm = take((size_t)BB * 128);
  float* dy_out      = take((size_t)M_dy * 128);    // sf_emb then lf_emb
  float* attn_d      = take((size_t)BT * NN);
  float* dyn_out     = take((size_t)BT * 128);
  float* bns_stats   = take(256);
  float* bnd_stats   = take(256);
  float* dyn_norm    = take((size_t)BT * 128);
  float* sfeat       = take((size_t)M1 * 128);
  float* stat_att    = take((size_t)BT * 128);
  float* dyn_att     = take((size_t)BT * 128);
  float* tf          = take((size_t)BT * 536);
  float* WihT0       = take((size_t)536 * 384);
  float* gi0         = take((size_t)BT * 384);

  auto cdiv = [](long a, long b) { return (int)((a + b - 1) / b); };

  // ---- 1. static MLP (sigmoid) over [station_nodes ; local_node] ----
  {
    GP g = { station_nodes, local_node, nullptr, nullptr, nullptr, nullptr };
    fused_mlp_kernel<0, 0, 16, 16><<<M_st / 16, 128, 0, stream>>>(g, M_st,
        st_W1, st_b1, st_W2, st_b2, st_lnw, st_lnb, st_out);
  }

  // ---- 2. static attention + BN over (B,1) ----
  attn_kernel<<<BB, 128, 0, stream>>>(st_out, st_out + (size_t)BB * NN * 128, attn_s, static_out);
  bn_stats_kernel<<<1, 128, 0, stream>>>(static_out, BB, bns_stats);
  bn_apply_kernel<<<BB, 128, 0, stream>>>(static_out, bns_stats, bn_s_w, bn_s_b, static_norm);

  // ---- 3. dynamic MLP (tanh) over station + local features (gather fused) ----
  {
    GP g = { station_features, local_features, weather_emb, wind_emb, station_emb, local_emb };
    fused_mlp_kernel<1, 1, 28, 27><<<M_dy / 16, 128, 0, stream>>>(g, M_dy,
        dy_W1, dy_b1, dy_W2, dy_b2, dy_lnw, dy_lnb, dy_out);
  }

  // ---- 4. dynamic attention + BN over (B,T,1) ----
  attn_kernel<<<BT, 128, 0, stream>>>(dy_out, dy_out + (size_t)M1 * 128, attn_d, dyn_out);
  bn_stats_kernel<<<1, 128, 0, stream>>>(dyn_out, BT, bnd_stats);
  bn_apply_kernel<<<BT, 128, 0, stream>>>(dyn_out, bnd_stats, bn_d_w, bn_d_b, dyn_norm);

  // ---- 5. station-fusion MLP (relu), gather fused ----
  {
    GP g = { st_out, dy_out, pm25_emb, nullptr, station_emb, nullptr };
    fused_mlp_kernel<2, 2, 264, 264><<<M1 / 16, 128, 0, stream>>>(g, M1,
        sf_W1, sf_b1, sf_W2, sf_b2, sf_lnw, sf_lnb, sfeat);
  }

  // ---- 6. attention-weighted combines ----
  att_combine_kernel<<<BT, 128, 0, stream>>>(sfeat, attn_s, attn_d, stat_att, dyn_att);

  // ---- 7. GRU input: concat + WMMA projection ----
  build_tf_kernel<<<cdiv((long)BT * 536, 128), 128, 0, stream>>>(
      static_norm, dyn_norm, stat_att, dyn_att, station_emb, month_emb, day_emb, hour_emb, tf);
  transpose_kernel<<<cdiv(384L * 536, 128), 128, 0, stream>>>(gru_Wih0, WihT0);
  gemm_bias_kernel<<<BT / 16, 128, 0, stream>>>(tf, BT, 536, WihT0, gru_bih0, 384, gi0);

  // ---- 8. sequential 2-layer GRU + tanh prediction head ----
  gru_kernel<<<BB, 128, 0, stream>>>(gi0, gru_Whh0, gru_bhh0, gru_Wih1, gru_bih1,
                                     gru_Whh1, gru_bhh1, pred_W, pred_b, out);
}